// MultiCrossEntropyLoss_55989193670921
// MI455X (gfx1250) — compile-verified
//
#include <hip/hip_runtime.h>
#include <cstdint>

// -----------------------------------------------------------------------------
// MultiCrossEntropyLoss for MI455X (gfx1250, wave32)
//   loss = -(1/N) * sum_i log(predicts[i, labels[i]]),  N = 1<<20, C = 128
//
// Bandwidth/latency-bound gather-reduce (~134MB effective HBM traffic ≈ 6us
// at 23.3 TB/s). CDNA5 features used:
//   * async global->LDS data mover (GLOBAL_LOAD_ASYNC_TO_LDS_B32, ASYNCcnt):
//     all 8 label chunks per wave issued back-to-back with shared base
//     registers + inst offsets (INST_OFFSET adds to BOTH the LDS and global
//     address, so LDS chunk stride is laid out to match the 128B global
//     stride), then a single s_wait_asynccnt 0.
//   * 8 scattered predicts gathers kept in flight per wave (no asm barriers
//     between them), wave32 butterfly reduction.
// WMMA intentionally unused: no matmul in this dataflow; a one-hot matmul
// would stream 512MB instead of ~134MB.
// -----------------------------------------------------------------------------

constexpr int kC            = 128;                        // classes (power of 2)
constexpr int kWaves        = 8;                          // waves per 256-thr block
constexpr int kChunk        = 32;                         // rows per wave-chunk (1/lane)
constexpr int kChunks       = 8;                          // chunks per wave
constexpr int kRowsPerBlock = kWaves * kChunk * kChunks;  // 2048

// Fast kernel: every row in [blockIdx*2048, blockIdx*2048+2048) is valid.
__global__ __launch_bounds__(256) void ce_partial_kernel(
    const float* __restrict__ predicts,
    const int*   __restrict__ labels,
    float*       __restrict__ partial)
{
    // slab[wave][chunk][lane]: chunk stride = 32*4B = 128B, matching the
    // global label stride per chunk so one inst-offset serves both addresses.
    __shared__ int slab[kWaves][kChunks][kChunk];         // 8KB

    const int lane = threadIdx.x & 31;
    const int wave = threadIdx.x >> 5;
    const long long waveBase =
        ((long long)blockIdx.x * kWaves + wave) * (long long)(kChunk * kChunks);

    // AS3 offset of this lane's chunk-0 slot (flat shared ptr low 32 bits ==
    // LDS byte offset, which is what the async mover's VDST VGPR expects).
    const unsigned ldsBase = (unsigned)(uintptr_t)&slab[wave][0][lane];
    const uint64_t gBase   = (uint64_t)(uintptr_t)(labels + waveBase + lane);

    // Issue all 8 coalesced async label loads; offsets are immediates.
#define CE_ASYNC_LD(OFF)                                                    \
    asm volatile("global_load_async_to_lds_b32 %0, %1, off offset:" #OFF    \
                 :: "v"(ldsBase), "v"(gBase) : "memory")
    CE_ASYNC_LD(0);   CE_ASYNC_LD(128); CE_ASYNC_LD(256); CE_ASYNC_LD(384);
    CE_ASYNC_LD(512); CE_ASYNC_LD(640); CE_ASYNC_LD(768); CE_ASYNC_LD(896);
#undef CE_ASYNC_LD
    asm volatile("s_wait_asynccnt 0x0" ::: "memory");

    // Per-lane row base: row(c) = waveBase + c*32 + lane
    //   -> element index = (waveBase+lane)*128 + c*4096 + lab
    const float* rb = predicts + (waveBase + (long long)lane) * kC;

    float acc0 = 0.0f, acc1 = 0.0f;
#pragma unroll
    for (int c = 0; c < kChunks; ++c) {
        const int lab = slab[wave][c][lane] & (kC - 1);     // ds_load_b32
        // Scattered non-temporal 4B gather; c*16384B folds into the imm
        // offset, so all 8 gathers stay in flight together.
        float p = __builtin_nontemporal_load(rb + c * (kChunk * kC) + lab);
        float lg = __logf(p);                               // v_log_f32 path
        if (c & 1) acc1 += lg; else acc0 += lg;
    }
    float acc = acc0 + acc1;

    // wave32 butterfly reduction
#pragma unroll
    for (int off = 16; off > 0; off >>= 1)
        acc += __shfl_xor(acc, off, 32);

    __shared__ float wsum[kWaves];
    if (lane == 0) wsum[wave] = acc;
    __syncthreads();
    if (threadIdx.x == 0) {
        float s = 0.0f;
#pragma unroll
        for (int w = 0; w < kWaves; ++w) s += wsum[w];
        partial[blockIdx.x] = s;
    }
}

// Tail kernel (generic n not divisible by 2048; never launched for N=1<<20).
__global__ __launch_bounds__(256) void ce_tail_kernel(
    const float* __restrict__ predicts,
    const int*   __restrict__ labels,
    float*       __restrict__ partialSlot,  // single output slot
    long long start, long long n)
{
    float acc = 0.0f;
    for (long long i = start + threadIdx.x; i < n; i += 256) {
        const int lab = labels[i] & (kC - 1);
        acc += __logf(predicts[i * kC + lab]);
    }
#pragma unroll
    for (int off = 16; off > 0; off >>= 1)
        acc += __shfl_xor(acc, off, 32);
    __shared__ float wsum[8];
    const int lane = threadIdx.x & 31, wave = threadIdx.x >> 5;
    if (lane == 0) wsum[wave] = acc;
    __syncthreads();
    if (threadIdx.x == 0) {
        float s = 0.0f;
#pragma unroll
        for (int w = 0; w < 8; ++w) s += wsum[w];
        partialSlot[0] = s;
    }
}

__global__ __launch_bounds__(256) void ce_final_kernel(
    const float* __restrict__ partial, int nb,
    float* __restrict__ out, float invN)
{
    __shared__ double sh[256];
    double s = 0.0;
    for (int i = threadIdx.x; i < nb; i += 256) s += (double)partial[i];
    sh[threadIdx.x] = s;
    __syncthreads();
#pragma unroll
    for (int stride = 128; stride > 0; stride >>= 1) {
        if ((int)threadIdx.x < stride) sh[threadIdx.x] += sh[threadIdx.x + stride];
        __syncthreads();
    }
    if (threadIdx.x == 0) out[0] = -(float)(sh[0] * (double)invN);
}

extern "C" void kernel_launch(void* const* d_in, const int* in_sizes, int n_in,
                              void* d_out, int out_size, void* d_ws, size_t ws_size,
                              hipStream_t stream) {
    const float* predicts = (const float*)d_in[0];   // [N,128] float32
    const int*   labels   = (const int*)d_in[1];     // [N] int (harness convention)
    const long long n = (long long)in_sizes[1];

    float* partial = (float*)d_ws;

    const int nbFull = (int)(n / kRowsPerBlock);     // 512 for N=1<<20
    const long long covered = (long long)nbFull * kRowsPerBlock;
    int nbTotal = nbFull;

    if (nbFull > 0)
        ce_partial_kernel<<<nbFull, 256, 0, stream>>>(predicts, labels, partial);
    if (covered < n) {
        ce_tail_kernel<<<1, 256, 0, stream>>>(predicts, labels,
                                              partial + nbFull, covered, n);
        nbTotal += 1;
    }
    ce_final_kernel<<<1, 256, 0, stream>>>(partial, nbTotal, (float*)d_out,
                                           1.0f / (float)n);
}